// GNNEncoder_26018911879756
// MI455X (gfx1250) — compile-verified
//
#include <hip/hip_runtime.h>
#include <hip/hip_bf16.h>

typedef __attribute__((ext_vector_type(2))) float v2f;
typedef __attribute__((ext_vector_type(8))) float v8f;

#define FDIM 64

// ---------------------------------------------------------------------------
// Zero a float buffer (agg sums / counts live in d_ws, which is poisoned).
// ---------------------------------------------------------------------------
__global__ void __launch_bounds__(256) zero_kernel(float* __restrict__ p, int n) {
    int i = blockIdx.x * 256 + threadIdx.x;
    if (i < n) p[i] = 0.0f;
}

// ---------------------------------------------------------------------------
// Edge scatter: 16 threads per edge, each handles 4 consecutive features.
// Coalesced float4 gather of feat[src], atomic f32 adds into agg[dst].
// Source features + accumulator both fit in the 192MB L2, so these atomics
// resolve at L2 without touching HBM.
// ---------------------------------------------------------------------------
__global__ void __launch_bounds__(256)
scatter_kernel(const float* __restrict__ feat, const int* __restrict__ src,
               const int* __restrict__ dst, float* __restrict__ agg,
               float* __restrict__ cnt, int n_edges) {
    long long tid = (long long)blockIdx.x * 256 + threadIdx.x;
    int e = (int)(tid >> 4);
    if (e >= n_edges) return;
    int q = (int)(tid & 15);
    int s = src[e];
    int d = dst[e];
    const float4 v = *(const float4*)(feat + (size_t)s * FDIM + q * 4);
    float* ap = agg + (size_t)d * FDIM + q * 4;
    atomicAdd(ap + 0, v.x);
    atomicAdd(ap + 1, v.y);
    atomicAdd(ap + 2, v.z);
    atomicAdd(ap + 3, v.w);
    if (cnt != nullptr && q == 0) atomicAdd(cnt + d, 1.0f);
}

// ---------------------------------------------------------------------------
// Fused SAGE transform:  out = [mean(agg)] @ Wl + xin @ Wr + bias  (opt. ReLU)
// One wave (32 lanes) computes a 16-node x 64-col tile via
// V_WMMA_F32_16X16X4_F32: K=64 -> 16 steps of K=4, 4 N-tiles of 16,
// two GEMMs accumulated into the same v8f accumulators. Weights staged in LDS.
// ---------------------------------------------------------------------------
__global__ void __launch_bounds__(256)
sage_gemm(const float* __restrict__ agg_sum, const float* __restrict__ cnt,
          const float* __restrict__ xin, const float* __restrict__ Wl,
          const float* __restrict__ Wr, const float* __restrict__ bias,
          float* __restrict__ out, int n_nodes, int relu_flag) {
    __shared__ float sWl[FDIM * FDIM];
    __shared__ float sWr[FDIM * FDIM];
    for (int i = threadIdx.x; i < FDIM * FDIM; i += 256) {
        sWl[i] = Wl[i];
        sWr[i] = Wr[i];
    }
    __syncthreads();

    const int lane = threadIdx.x & 31;
    const int wave = threadIdx.x >> 5;
    const int tile = blockIdx.x * 8 + wave;
    const int ntiles = (n_nodes + 15) >> 4;
    if (tile >= ntiles) return;  // wave-uniform: EXEC stays all-ones for WMMA
    const int node0 = tile << 4;

    const int half = lane >> 4;   // K sub-pair selector for A/B fragments
    const int lm   = lane & 15;   // A-row / B-col / C-col within the tile

    // Mean scale for the aggregate rows (A rows indexed by lm).
    float c   = (cnt != nullptr) ? cnt[node0 + lm] : 1.0f;
    float inv = 1.0f / fmaxf(c, 1.0f);

    const float* arow = agg_sum + (size_t)(node0 + lm) * FDIM;
    const float* xrow = xin     + (size_t)(node0 + lm) * FDIM;

    v8f acc[4];
    #pragma unroll
    for (int nt = 0; nt < 4; ++nt) acc[nt] = (v8f){};

    // GEMM 1: mean(agg) @ Wl
    #pragma unroll
    for (int kb = 0; kb < 16; ++kb) {
        const int k0 = kb * 4 + half * 2;
        v2f a;
        a.x = arow[k0]     * inv;
        a.y = arow[k0 + 1] * inv;
        #pragma unroll
        for (int nt = 0; nt < 4; ++nt) {
            v2f b;
            b.x = sWl[k0 * FDIM + nt * 16 + lm];
            b.y = sWl[(k0 + 1) * FDIM + nt * 16 + lm];
            acc[nt] = __builtin_amdgcn_wmma_f32_16x16x4_f32(
                false, a, false, b, (short)0, acc[nt], false, false);
        }
    }
    // GEMM 2: xin @ Wr (accumulate)
    #pragma unroll
    for (int kb = 0; kb < 16; ++kb) {
        const int k0 = kb * 4 + half * 2;
        v2f a;
        a.x = xrow[k0];
        a.y = xrow[k0 + 1];
        #pragma unroll
        for (int nt = 0; nt < 4; ++nt) {
            v2f b;
            b.x = sWr[k0 * FDIM + nt * 16 + lm];
            b.y = sWr[(k0 + 1) * FDIM + nt * 16 + lm];
            acc[nt] = __builtin_amdgcn_wmma_f32_16x16x4_f32(
                false, a, false, b, (short)0, acc[nt], false, false);
        }
    }

    // Bias + optional ReLU + store.
    // C/D layout: VGPR i holds rows {i, i+8} (by half-wave), col = lm.
    #pragma unroll
    for (int nt = 0; nt < 4; ++nt) {
        const float bv = bias[nt * 16 + lm];
        #pragma unroll
        for (int i = 0; i < 8; ++i) {
            float v = acc[nt][i] + bv;
            if (relu_flag) v = fmaxf(v, 0.0f);
            out[(size_t)(node0 + i + half * 8) * FDIM + nt * 16 + lm] = v;
        }
    }
}

// ---------------------------------------------------------------------------
// Host-side launch sequence (graph-capture safe: only kernel launches).
// ---------------------------------------------------------------------------
extern "C" void kernel_launch(void* const* d_in, const int* in_sizes, int n_in,
                              void* d_out, int out_size, void* d_ws, size_t ws_size,
                              hipStream_t stream) {
    const float* x    = (const float*)d_in[0];
    const int*   ei   = (const int*)d_in[1];
    const float* W1_l = (const float*)d_in[2];
    const float* b1   = (const float*)d_in[3];
    const float* W1_r = (const float*)d_in[4];
    const float* W2_l = (const float*)d_in[5];
    const float* b2   = (const float*)d_in[6];
    const float* W2_r = (const float*)d_in[7];
    float* out = (float*)d_out;

    const int n_nodes = in_sizes[0] / FDIM;
    const int n_edges = in_sizes[1] / 2;
    const int* src = ei;
    const int* dst = ei + n_edges;

    // Workspace layout: agg[N*64] | cnt[N] | h[N*64]
    float* agg = (float*)d_ws;
    float* cnt = agg + (size_t)n_nodes * FDIM;
    float* h   = cnt + n_nodes;

    const int ntiles     = (n_nodes + 15) / 16;
    const int gemm_grid  = (ntiles + 7) / 8;
    const long long sthreads = (long long)n_edges * 16;
    const int scat_grid  = (int)((sthreads + 255) / 256);
    const int nagg       = n_nodes * FDIM;

    // ---- Layer 1 ----
    zero_kernel<<<(nagg + n_nodes + 255) / 256, 256, 0, stream>>>(agg, nagg + n_nodes);
    scatter_kernel<<<scat_grid, 256, 0, stream>>>(x, src, dst, agg, cnt, n_edges);
    sage_gemm<<<gemm_grid, 256, 0, stream>>>(agg, cnt, x, W1_l, W1_r, b1, h,
                                             n_nodes, /*relu=*/1);

    // ---- Layer 2 (degree counts reused) ----
    zero_kernel<<<(nagg + 255) / 256, 256, 0, stream>>>(agg, nagg);
    scatter_kernel<<<scat_grid, 256, 0, stream>>>(h, src, dst, agg, nullptr, n_edges);
    sage_gemm<<<gemm_grid, 256, 0, stream>>>(agg, cnt, h, W2_l, W2_r, b2, out,
                                             n_nodes, /*relu=*/0);
}